// ANIRepresentation_10977936408846
// MI455X (gfx1250) — compile-verified
//
#include <hip/hip_runtime.h>
#include <hip/hip_bf16.h>
#include <stdint.h>

// ---------------------------------------------------------------------------
// ANI symmetry functions for MI455X (gfx1250, wave32).
//   out[0 .. nAtoms*7*8)        = radial AEV (HW f32 atomic scatter-add, L2-resident)
//   out[nAtoms*56 .. +T*32)     = angular features (LDS-staged -> async DMA, NT)
// ---------------------------------------------------------------------------

#define PI_F 3.14159265358979323846f

typedef float v4f __attribute__((ext_vector_type(4)));

// --- guaranteed HW float atomic (no CAS fallback), no-return form ----------
__device__ __forceinline__ void gatomic_fadd(float* p, float v) {
  asm volatile("global_atomic_add_f32 %0, %1, off"
               :: "v"((unsigned long long)(uintptr_t)p), "v"(v)
               : "memory");
}

// --- gfx1250 async store-from-LDS (ASYNCcnt path), non-temporal ------------
__device__ __forceinline__ void async_store_from_lds_b128_nt(unsigned long long gaddr,
                                                             unsigned ldsoff) {
  // VFLAT async store: vaddr(64b pair), vsrc = per-lane LDS byte address
  asm volatile("global_store_async_from_lds_b128 %0, %1, off th:TH_STORE_NT"
               :: "v"(gaddr), "v"(ldsoff)
               : "memory");
}

typedef __attribute__((address_space(3))) void lds_void;

// ---------------------------------------------------------------------------
// Kernel 1: zero the AEV accumulation region (atomics accumulate each launch).
// Regular-temporal: this region stays hot in L2 for the atomic scatter.
// ---------------------------------------------------------------------------
__global__ void __launch_bounds__(256) ani_zero_f4(v4f* __restrict__ p, int n4) {
  int i = blockIdx.x * blockDim.x + threadIdx.x;
  if (i < n4) p[i] = (v4f){0.f, 0.f, 0.f, 0.f};
}

// ---------------------------------------------------------------------------
// Kernel 2: radial branch. One thread per (pair, k): 8 consecutive lanes hit
// 8 consecutive floats -> atomics coalesce into 32B L2 transactions.
// ---------------------------------------------------------------------------
__global__ void __launch_bounds__(256) ani_radial(
    const float* __restrict__ r_ij, const int* __restrict__ pairs,
    const int* __restrict__ Z, float* __restrict__ aev, int P) {
  int tid = blockIdx.x * blockDim.x + threadIdx.x;
  int p = tid >> 3, k = tid & 7;
  if (p >= P) return;

  float d  = r_ij[p];
  int   i0 = pairs[p];
  int   i1 = pairs[P + p];
  int   s0 = Z[i0];
  int   s1 = Z[i1];

  // cosine cutoff, rc = 5.1
  float fc = (d <= 5.1f) ? (__cosf(d * (PI_F / 5.1f)) * 0.5f + 0.5f) : 0.f;
  // ShfR[k] = 0.8 + (4.3/8)*k
  float x = d - (0.8f + 0.5375f * (float)k);
  float g = 0.25f * fc * __expf(-19.7f * x * x);

  // index12 = idx*7 + partner_species ; row length 8
  gatomic_fadd(&aev[(i0 * 7 + s1) * 8 + k], g);
  gatomic_fadd(&aev[(i1 * 7 + s0) * 8 + k], g);
}

// ---------------------------------------------------------------------------
// Kernel 3: angular branch. Each lane computes one triplet's 32-float row.
// Rows are staged into a wave-private LDS buffer (double-buffered) and DMA'd
// out as linear 512B chunks with global_store_async_from_lds_b128 (NT).
// ---------------------------------------------------------------------------
__global__ void __launch_bounds__(128) ani_angular(
    const float* __restrict__ vec12, float* __restrict__ ang, int T) {
  // 4 waves * 2 buffers * 1024 floats(4KB) = 32KB static LDS
  __shared__ __align__(16) float smem[4 * 2 * 1024];

  const int lane = threadIdx.x & 31;
  const int wib  = threadIdx.x >> 5;
  const long long gwave = (long long)blockIdx.x * 4 + wib;
  const long long nwave = (long long)gridDim.x * 4;
  const unsigned smemBase = (unsigned)(unsigned long long)(lds_void*)smem;

  // cos/sin of ShfZ[a] = (a+0.5)*pi/4
  const float cZ0 =  0.92387953f, cZ1 =  0.38268343f, cZ2 = -0.38268343f, cZ3 = -0.92387953f;
  const float sZ0 =  0.38268343f, sZ1 =  0.92387953f, sZ2 =  0.92387953f, sZ3 =  0.38268343f;

  int it = 0;
  for (long long tile = gwave; tile * 32 < (long long)T; tile += nwave, ++it) {
    const long long t0 = tile * 32;
    const long long t  = t0 + lane;
    const bool valid = (t < (long long)T);

    float out[32];
    if (valid) {
      const float* v1 = vec12 + (size_t)t * 3;
      const float* v2 = vec12 + (size_t)T * 3 + (size_t)t * 3;
      // streaming read-once inputs: non-temporal loads
      float ax = __builtin_nontemporal_load(v1 + 0);
      float ay = __builtin_nontemporal_load(v1 + 1);
      float az = __builtin_nontemporal_load(v1 + 2);
      float bx = __builtin_nontemporal_load(v2 + 0);
      float by = __builtin_nontemporal_load(v2 + 1);
      float bz = __builtin_nontemporal_load(v2 + 2);

      float da2 = ax * ax + ay * ay + az * az;
      float db2 = bx * bx + by * by + bz * bz;
      float dab = ax * bx + ay * by + az * bz;

      float d1 = __fsqrt_rn(da2);
      float d2 = __fsqrt_rn(db2);
      float ct = 0.95f * dab * __frsqrt_rn(da2 * db2);     // cos(theta)
      float st = __fsqrt_rn(fmaxf(1.f - ct * ct, 0.f));    // sin(theta) >= 0

      float fc1 = (d1 <= 3.5f) ? (__cosf(d1 * (PI_F / 3.5f)) * 0.5f + 0.5f) : 0.f;
      float fc2 = (d2 <= 3.5f) ? (__cosf(d2 * (PI_F / 3.5f)) * 0.5f + 0.5f) : 0.f;
      float fcp = 2.f * fc1 * fc2;

      float avg = 0.5f * (d1 + d2);
      float fr[8];
#pragma unroll
      for (int k = 0; k < 8; ++k) {
        float x = avg - (0.8f + 0.3375f * (float)k);       // ShfA
        fr[k] = __expf(-12.5f * x * x);
      }

      // f_ang[a] = ((1 + cos(theta - ShfZ[a]))/2)^14.1, via angle-sum identity
      float fa[4];
      {
        float c0 = ct * cZ0 + st * sZ0;
        float c1 = ct * cZ1 + st * sZ1;
        float c2 = ct * cZ2 + st * sZ2;
        float c3 = ct * cZ3 + st * sZ3;
        fa[0] = fcp * __powf(0.5f + 0.5f * c0, 14.1f);
        fa[1] = fcp * __powf(0.5f + 0.5f * c1, 14.1f);
        fa[2] = fcp * __powf(0.5f + 0.5f * c2, 14.1f);
        fa[3] = fcp * __powf(0.5f + 0.5f * c3, 14.1f);
      }
#pragma unroll
      for (int a = 0; a < 4; ++a)
#pragma unroll
        for (int k = 0; k < 8; ++k) out[a * 8 + k] = fa[a] * fr[k];
    }

    if (t0 + 32 <= (long long)T) {
      // ---- full tile: stage in LDS, DMA out linearly (ASYNCcnt path) ----
      // WAR guard: allow only the previous parity's 8 async stores in flight.
      asm volatile("s_wait_asynccnt 8" ::: "memory");

      const int bufIdx = wib * 2 + (it & 1);
      v4f* slot = (v4f*)(smem + bufIdx * 1024 + lane * 32);
#pragma unroll
      for (int j = 0; j < 8; ++j)
        slot[j] = (v4f){out[4 * j], out[4 * j + 1], out[4 * j + 2], out[4 * j + 3]};

      // LDS writes must land before the async engine reads them.
      asm volatile("s_wait_dscnt 0" ::: "memory");

      const unsigned ldsBuf = smemBase + (unsigned)(bufIdx * 4096);
      const unsigned long long gbase =
          (unsigned long long)(uintptr_t)ang + (unsigned long long)t0 * 128ull;
#pragma unroll
      for (int i = 0; i < 8; ++i) {
        unsigned off = (unsigned)(i * 512 + lane * 16);
        async_store_from_lds_b128_nt(gbase + off, ldsBuf + off);
      }
      // trailing async stores drained by s_endpgm's implicit wait-idle
    } else if (valid) {
      // ---- tail tile: direct guarded non-temporal stores ----
      v4f* o = (v4f*)(ang + (size_t)t * 32);
#pragma unroll
      for (int j = 0; j < 8; ++j) {
        v4f v = (v4f){out[4 * j], out[4 * j + 1], out[4 * j + 2], out[4 * j + 3]};
        __builtin_nontemporal_store(v, o + j);
      }
    }
  }
}

// ---------------------------------------------------------------------------
extern "C" void kernel_launch(void* const* d_in, const int* in_sizes, int n_in,
                              void* d_out, int out_size, void* d_ws, size_t ws_size,
                              hipStream_t stream) {
  (void)n_in; (void)out_size; (void)d_ws; (void)ws_size;

  const float* r_ij  = (const float*)d_in[0];
  const int*   pairs = (const int*)d_in[1];
  const int*   Z     = (const int*)d_in[2];
  const float* vec12 = (const float*)d_in[3];

  const int P      = in_sizes[0];        // 1,600,000 pairs
  const int nAtoms = in_sizes[2];        // 50,000
  const int T      = in_sizes[3] / 6;    // 1,600,000 triplets

  float* aev = (float*)d_out;                          // [nAtoms*7, 8]
  float* ang = (float*)d_out + (size_t)nAtoms * 56;    // [T, 32]

  // 1) zero AEV region (nAtoms*56 is divisible by 8 -> by 4)
  const int n4 = (nAtoms * 56) / 4;
  ani_zero_f4<<<(n4 + 255) / 256, 256, 0, stream>>>((v4f*)aev, n4);

  // 2) radial scatter (thread per (pair, feature))
  const long long nr = (long long)P * 8;
  ani_radial<<<(unsigned)((nr + 255) / 256), 256, 0, stream>>>(r_ij, pairs, Z, aev, P);

  // 3) angular streaming (grid-stride; ~5 tiles/wave to realize double buffering)
  const int tiles = (T + 31) / 32;
  int angBlocks = (tiles + 3) / 4;
  if (angBlocks > 2500) angBlocks = 2500;
  if (angBlocks < 1) angBlocks = 1;
  ani_angular<<<angBlocks, 128, 0, stream>>>(vec12, ang, T);
}